// ColorGuidedFilter_50440095924829
// MI455X (gfx1250) — compile-verified
//
#include <hip/hip_runtime.h>
#include <stdint.h>

// ---------------- problem constants ----------------
#define GF_B 4
#define GF_C 3
#define GF_H 1024
#define GF_W 1024
#define GF_R 8
#define GF_D 17                      // 2R+1
#define GF_TW 32
#define GF_TH 32
#define GF_PW (GF_TW + 2*GF_R)       // 48
#define GF_PH (GF_TH + 2*GF_R)       // 48
#define GF_NPAD (GF_PH*GF_PW)        // 2304 staged elements per tile
#define NTHREADS 256
#define GF_EPS 0.01f
#define GF_INV_AREA (1.0f/289.0f)

// ---------------- CDNA5 async global->LDS path ----------------
#if defined(__has_builtin)
#  if __has_builtin(__builtin_amdgcn_global_load_async_to_lds_b32)
#    define GF_ASYNC 1
#  endif
#endif

typedef __attribute__((address_space(1))) int gf_gint;
typedef __attribute__((address_space(3))) int gf_lint;

// Async copy of one f32, with a compile-time byte offset applied to both the
// global source and the LDS destination (ISA: INST_OFFSET is added to both).
template<int OFF>
__device__ __forceinline__ void gf_copy_f32_off(const float* __restrict__ g, float* l) {
#ifdef GF_ASYNC
  __builtin_amdgcn_global_load_async_to_lds_b32(
      (gf_gint*)(uintptr_t)(const void*)g,
      (gf_lint*)(uintptr_t)(void*)l,
      OFF, 0);
#else
  *(float*)((char*)l + OFF) = *(const float*)((const char*)g + OFF);
#endif
}

__device__ __forceinline__ void gf_copy_f32(const float* __restrict__ g, float* l) {
  gf_copy_f32_off<0>(g, l);
}

__device__ __forceinline__ void gf_async_wait() {
#ifdef GF_ASYNC
#  if __has_builtin(__builtin_amdgcn_s_wait_asynccnt)
  __builtin_amdgcn_s_wait_asynccnt(0);
#  else
  asm volatile("s_wait_asynccnt 0" ::: "memory");
#  endif
#endif
}

__device__ __forceinline__ int reflect_i(int i, int n) {
  i = (i < 0) ? -i : i;              // jnp.pad 'reflect': -k -> k
  i = (i >= n) ? (2*n - 2 - i) : i;  // n-1+k -> n-1-k
  return i;
}

// =====================================================================
// Kernel 1: box-filter 13 channel products, 3x3 solve -> AB = (a0,a1,a2,b)
// =====================================================================
__global__ __launch_bounds__(NTHREADS)
void gf_stats_kernel(const float* __restrict__ I, const float* __restrict__ p,
                     float4* __restrict__ AB) {
  __shared__ float sI0[GF_NPAD];
  __shared__ float sI1[GF_NPAD];
  __shared__ float sI2[GF_NPAD];
  __shared__ float sp [GF_NPAD];
  __shared__ float hs [GF_PH*GF_TW];   // horizontal sums scratch

  const int tid = threadIdx.x;
  const int x0 = blockIdx.x * GF_TW;
  const int y0 = blockIdx.y * GF_TH;
  const int bz = blockIdx.z;

  const size_t plane = (size_t)GF_H * GF_W;
  const float* Ib = I + (size_t)bz * GF_C * plane;
  const float* pb = p + (size_t)bz * plane;

  // ---- stage 4 channels with reflect halo (async -> LDS) ----
  for (int it = tid; it < GF_NPAD; it += NTHREADS) {   // 2304/256 = 9 iters
    int ly = it / GF_PW, lx = it % GF_PW;
    int gy = reflect_i(y0 + ly - GF_R, GF_H);
    int gx = reflect_i(x0 + lx - GF_R, GF_W);
    size_t gidx = (size_t)gy * GF_W + gx;
    gf_copy_f32(Ib + 0*plane + gidx, &sI0[it]);
    gf_copy_f32(Ib + 1*plane + gidx, &sI1[it]);
    gf_copy_f32(Ib + 2*plane + gidx, &sI2[it]);
    gf_copy_f32(pb + gidx,           &sp [it]);
  }
  gf_async_wait();
  __syncthreads();

  // means: 0..2 = mean_I, 3 = mean_p, 4..6 = mean_Ip, 7..12 = mean_II(00,01,02,11,12,22)
  float mean[13][4];

#pragma unroll
  for (int ch = 0; ch < 13; ++ch) {
    // phase 1: horizontal 17-tap sums over all 48 rows x 32 cols
    for (int it = tid; it < GF_PH*GF_TW; it += NTHREADS) { // 1536/256 = 6 iters
      int y = it / GF_TW, x = it % GF_TW;
      int base = y * GF_PW + x;
      float s = 0.f;
#pragma unroll
      for (int dx = 0; dx < GF_D; ++dx) {
        int j = base + dx;
        float v;
        if      (ch == 0)  v = sI0[j];
        else if (ch == 1)  v = sI1[j];
        else if (ch == 2)  v = sI2[j];
        else if (ch == 3)  v = sp [j];
        else if (ch == 4)  v = sI0[j]*sp [j];
        else if (ch == 5)  v = sI1[j]*sp [j];
        else if (ch == 6)  v = sI2[j]*sp [j];
        else if (ch == 7)  v = sI0[j]*sI0[j];
        else if (ch == 8)  v = sI0[j]*sI1[j];
        else if (ch == 9)  v = sI0[j]*sI2[j];
        else if (ch == 10) v = sI1[j]*sI1[j];
        else if (ch == 11) v = sI1[j]*sI2[j];
        else               v = sI2[j]*sI2[j];
        s += v;
      }
      hs[it] = s;
    }
    __syncthreads();
    // phase 2: vertical 17-tap sums -> registers
#pragma unroll
    for (int k = 0; k < 4; ++k) {
      int it = tid + k*NTHREADS;
      int y = it / GF_TW, x = it % GF_TW;
      float s = 0.f;
#pragma unroll
      for (int dy = 0; dy < GF_D; ++dy) s += hs[(y+dy)*GF_TW + x];
      mean[ch][k] = s * GF_INV_AREA;
    }
    __syncthreads();   // hs is rewritten next channel
  }

  // ---- per-pixel 3x3 symmetric solve (Cramer), packed b128 store ----
  float4* ABb = AB + (size_t)bz * plane;
#pragma unroll
  for (int k = 0; k < 4; ++k) {
    int it = tid + k*NTHREADS;
    int y = it / GF_TW, x = it % GF_TW;
    float mI0 = mean[0][k], mI1 = mean[1][k], mI2 = mean[2][k];
    float mp  = mean[3][k];
    float c0 = mean[4][k] - mI0*mp;
    float c1 = mean[5][k] - mI1*mp;
    float c2 = mean[6][k] - mI2*mp;
    float S00 = mean[7][k]  - mI0*mI0 + GF_EPS;
    float S01 = mean[8][k]  - mI0*mI1;
    float S02 = mean[9][k]  - mI0*mI2;
    float S11 = mean[10][k] - mI1*mI1 + GF_EPS;
    float S12 = mean[11][k] - mI1*mI2;
    float S22 = mean[12][k] - mI2*mI2 + GF_EPS;
    float A00 = S11*S22 - S12*S12;
    float A01 = S02*S12 - S01*S22;
    float A02 = S01*S12 - S02*S11;
    float A11 = S00*S22 - S02*S02;
    float A12 = S01*S02 - S00*S12;
    float A22 = S00*S11 - S01*S01;
    float det = S00*A00 + S01*A01 + S02*A02;
    float rdet = 1.0f / det;
    float a0 = (A00*c0 + A01*c1 + A02*c2) * rdet;
    float a1 = (A01*c0 + A11*c1 + A12*c2) * rdet;
    float a2 = (A02*c0 + A12*c1 + A22*c2) * rdet;
    float bb = mp - (a0*mI0 + a1*mI1 + a2*mI2);
    size_t pix = (size_t)(y0 + y) * GF_W + (x0 + x);
    ABb[pix] = make_float4(a0, a1, a2, bb);   // coalesced global_store_b128
  }
}

// =====================================================================
// Kernel 2: box-filter (a0,a1,a2,b), q = mean_a . I + mean_b
// LDS tile is AoS with stride 5 (4 data + 1 pad) -> conflict-free banks.
// =====================================================================
__global__ __launch_bounds__(NTHREADS)
void gf_output_kernel(const float* __restrict__ I, const float4* __restrict__ AB,
                      float* __restrict__ q) {
  __shared__ float sab[GF_NPAD * 5];
  __shared__ float hs [GF_PH*GF_TW];

  const int tid = threadIdx.x;
  const int x0 = blockIdx.x * GF_TW;
  const int y0 = blockIdx.y * GF_TH;
  const int bz = blockIdx.z;

  const size_t plane = (size_t)GF_H * GF_W;
  const float4* ABb = AB + (size_t)bz * plane;

  // ---- stage (a0,a1,a2,b) quads with reflect halo ----
  for (int it = tid; it < GF_NPAD; it += NTHREADS) {
    int ly = it / GF_PW, lx = it % GF_PW;
    int gy = reflect_i(y0 + ly - GF_R, GF_H);
    int gx = reflect_i(x0 + lx - GF_R, GF_W);
    size_t gidx = (size_t)gy * GF_W + gx;
    const float* gp = (const float*)(ABb + gidx);   // 16B contiguous quad
    float* lp = &sab[it * 5];                       // 4 contiguous + pad
    gf_copy_f32_off<0> (gp, lp);
    gf_copy_f32_off<4> (gp, lp);
    gf_copy_f32_off<8> (gp, lp);
    gf_copy_f32_off<12>(gp, lp);
  }
  gf_async_wait();
  __syncthreads();

  float mean[4][4];
#pragma unroll
  for (int ch = 0; ch < 4; ++ch) {
    for (int it = tid; it < GF_PH*GF_TW; it += NTHREADS) {
      int y = it / GF_TW, x = it % GF_TW;
      int base = y * GF_PW + x;
      float s = 0.f;
#pragma unroll
      for (int dx = 0; dx < GF_D; ++dx) {
        s += sab[(base + dx) * 5 + ch];
      }
      hs[it] = s;
    }
    __syncthreads();
#pragma unroll
    for (int k = 0; k < 4; ++k) {
      int it = tid + k*NTHREADS;
      int y = it / GF_TW, x = it % GF_TW;
      float s = 0.f;
#pragma unroll
      for (int dy = 0; dy < GF_D; ++dy) s += hs[(y+dy)*GF_TW + x];
      mean[ch][k] = s * GF_INV_AREA;
    }
    __syncthreads();
  }

  const float* Ib = I + (size_t)bz * GF_C * plane;
  float* qb = q + (size_t)bz * plane;
#pragma unroll
  for (int k = 0; k < 4; ++k) {
    int it = tid + k*NTHREADS;
    int y = it / GF_TW, x = it % GF_TW;
    size_t pix = (size_t)(y0 + y) * GF_W + (x0 + x);
    float qv = mean[0][k] * Ib[0*plane + pix]
             + mean[1][k] * Ib[1*plane + pix]
             + mean[2][k] * Ib[2*plane + pix]
             + mean[3][k];
    __builtin_nontemporal_store(qv, &qb[pix]);   // write-once output
  }
}

// =====================================================================
extern "C" void kernel_launch(void* const* d_in, const int* in_sizes, int n_in,
                              void* d_out, int out_size, void* d_ws, size_t ws_size,
                              hipStream_t stream) {
  (void)in_sizes; (void)n_in; (void)out_size; (void)ws_size;
  const float* I = (const float*)d_in[0];   // (4,3,1024,1024) f32
  const float* p = (const float*)d_in[1];   // (4,1,1024,1024) f32
  float* q = (float*)d_out;                 // (4,1,1024,1024) f32

  // workspace: AB[b][pix] = (a0,a1,a2,b) as float4 -> 64 MB total
  float4* AB = (float4*)d_ws;

  dim3 grid(GF_W/GF_TW, GF_H/GF_TH, GF_B);
  gf_stats_kernel <<<grid, NTHREADS, 0, stream>>>(I, p, AB);
  gf_output_kernel<<<grid, NTHREADS, 0, stream>>>(I, AB, q);
}